// GreedyCTCDecoder_57329223467712
// MI455X (gfx1250) — compile-verified
//
#include <hip/hip_runtime.h>
#include <hip/hip_bf16.h>

// GreedyCTCDecoder for MI455X (gfx1250, wave32).
// Layout: d_in[0] = emission [64, 8192, 32] fp32
//         d_out   = btokens [64, 8192] int32  ++  lens [64] int32
//         d_ws    = idx u8 [64*8192]  ++  col_keep u8 [8192]

#define CTC_B 64
#define CTC_T 8192
#define CTC_L 32
#define CTC_BLANK 0

// LDS row stride in bytes: 32 floats (128 B) + 8 B pad.
// 136 B = 34 dwords; 34*lane mod 64 is a bijection over lanes 0..31 -> the
// per-thread float2 row reads are bank-conflict free, and every address is
// 8-byte aligned (matches the b64 async deposit granularity).
#define ROWSTRIDE 136

#if __has_builtin(__builtin_amdgcn_global_load_async_to_lds_b64)
#define USE_ASYNC_LDS 1
#else
#define USE_ASYNC_LDS 0
#endif

// Exact pointee type the async builtin expects: <2 x i32> ("b64").
typedef int ctc_v2i __attribute__((ext_vector_type(2)));
#define GBL_V2I(p) ((__attribute__((address_space(1))) ctc_v2i*)(p))
#define LDS_V2I(p) ((__attribute__((address_space(3))) ctc_v2i*)(p))

static __device__ __forceinline__ void wait_asynccnt0() {
#if __has_builtin(__builtin_amdgcn_s_wait_asynccnt)
    __builtin_amdgcn_s_wait_asynccnt(0);
#else
    asm volatile("s_wait_asynccnt 0" ::: "memory");
#endif
}

// ---------------------------------------------------------------------------
// Kernel 1: per-(b,t) argmax over 32 labels. One thread per position.
// Each 256-thread block stages a 32 KB tile (256 rows x 128 B) into LDS via
// async global->LDS copies (ASYNCcnt), then reduces from LDS.
// ---------------------------------------------------------------------------
__global__ __launch_bounds__(256) void ctc_argmax_kernel(
    const float* __restrict__ emission, unsigned char* __restrict__ idx_out) {
    __shared__ unsigned char smem[256 * ROWSTRIDE];

    const int tid = threadIdx.x;
    const long long g0 = (long long)blockIdx.x * 256;          // first position
    const unsigned char* gbase =
        (const unsigned char*)emission + g0 * (CTC_L * 4);     // 128 B / pos

#if USE_ASYNC_LDS
    // 16 iterations x 256 threads x 8 B = 32 KB, fully coalesced 2 KB bursts.
#pragma unroll
    for (int k = 0; k < 16; ++k) {
        const int off = (k * 256 + tid) * 8;                   // linear byte
        const int row = off >> 7;                              // / 128
        const int col = off & 127;
        __builtin_amdgcn_global_load_async_to_lds_b64(
            GBL_V2I(gbase + off),
            LDS_V2I(smem + row * ROWSTRIDE + col),
            /*offset=*/0, /*cpol=*/0);
    }
    wait_asynccnt0();
    __syncthreads();
#else
    // Fallback: coalesced global loads + DS stores, same padded layout.
#pragma unroll
    for (int k = 0; k < 16; ++k) {
        const int off = (k * 256 + tid) * 8;
        const int row = off >> 7;
        const int col = off & 127;
        *(uint2*)(smem + row * ROWSTRIDE + col) =
            *(const uint2*)(gbase + off);
    }
    __syncthreads();
#endif

    // Thread tid reduces its own 128 B row (bank-conflict-free float2 reads).
    const float2* rowp = (const float2*)(smem + tid * ROWSTRIDE);
    float2 v0 = rowp[0];
    float best = v0.x;
    int arg = 0;
    if (v0.y > best) { best = v0.y; arg = 1; }
#pragma unroll
    for (int j = 1; j < 16; ++j) {
        float2 v = rowp[j];
        if (v.x > best) { best = v.x; arg = 2 * j; }
        if (v.y > best) { best = v.y; arg = 2 * j + 1; }
    }
    idx_out[g0 + tid] = (unsigned char)arg;   // labels < 32 fit in u8
}

// ---------------------------------------------------------------------------
// Kernel 2: col_keep[t] = (t==0) || any_b(idx[b,t] != idx[b,t-1]).
// t is the fast index across threads -> every load is coalesced; ~1 MiB of
// L2-resident traffic.
// ---------------------------------------------------------------------------
__global__ __launch_bounds__(256) void ctc_colkeep_kernel(
    const unsigned char* __restrict__ idx, unsigned char* __restrict__ col_keep) {
    const int t = blockIdx.x * 256 + threadIdx.x;
    if (t >= CTC_T) return;
    int keep;
    if (t == 0) {
        keep = 1;
    } else {
        keep = 0;
        for (int b = 0; b < CTC_B; ++b) {
            const unsigned char* row = idx + (size_t)b * CTC_T;
            if (row[t] != row[t - 1]) { keep = 1; break; }
        }
    }
    col_keep[t] = (unsigned char)keep;
}

// ---------------------------------------------------------------------------
// Kernel 3: per-row stream compaction. One 256-thread block (8 waves) per
// batch row; 32 sequential chunks with a ballot-based block scan (wave32).
// Writes kept tokens left-packed, zero-fills the tail, emits lens[b].
// ---------------------------------------------------------------------------
__global__ __launch_bounds__(256) void ctc_compact_kernel(
    const unsigned char* __restrict__ idx,
    const unsigned char* __restrict__ col_keep,
    int* __restrict__ out) {
    __shared__ int waveCnt[8];

    const int b = blockIdx.x;
    const int tid = threadIdx.x;
    const int lane = tid & 31;
    const int wave = tid >> 5;
    const unsigned char* rowp = idx + (size_t)b * CTC_T;
    int* outp = out + (size_t)b * CTC_T;

    int running = 0;  // identical across all threads of the block
    for (int c = 0; c < CTC_T / 256; ++c) {
        const int t = c * 256 + tid;
        const int v = rowp[t];
        const bool keep = (col_keep[t] != 0) && (v != CTC_BLANK);

        const unsigned mask = __builtin_amdgcn_ballot_w32(keep);
        const int prefix = __builtin_popcount(mask & ((1u << lane) - 1u));
        if (lane == 0) waveCnt[wave] = __builtin_popcount(mask);
        __syncthreads();

        int woff = 0, ctot = 0;
#pragma unroll
        for (int w = 0; w < 8; ++w) {
            const int cw = waveCnt[w];
            if (w < wave) woff += cw;
            ctot += cw;
        }
        if (keep) outp[running + woff + prefix] = v;
        running += ctot;
        __syncthreads();  // protect waveCnt before next chunk overwrites it
    }

    // Blank-pad the tail so every output element is written each launch.
    for (int p = running + tid; p < CTC_T; p += 256) outp[p] = CTC_BLANK;
    if (tid == 0) out[(size_t)CTC_B * CTC_T + b] = running;
}

// ---------------------------------------------------------------------------
extern "C" void kernel_launch(void* const* d_in, const int* in_sizes, int n_in,
                              void* d_out, int out_size, void* d_ws, size_t ws_size,
                              hipStream_t stream) {
    (void)in_sizes; (void)n_in; (void)out_size; (void)ws_size;
    const float* emission = (const float*)d_in[0];
    int* out = (int*)d_out;  // int32 outputs: btokens [B*T] then lens [B]

    unsigned char* idx = (unsigned char*)d_ws;                 // B*T bytes
    unsigned char* col_keep = idx + (size_t)CTC_B * CTC_T;     // T bytes

    ctc_argmax_kernel<<<(CTC_B * CTC_T) / 256, 256, 0, stream>>>(emission, idx);
    ctc_colkeep_kernel<<<CTC_T / 256, 256, 0, stream>>>(idx, col_keep);
    ctc_compact_kernel<<<CTC_B, 256, 0, stream>>>(idx, col_keep, out);
}